// GraphConv_87866440942242
// MI455X (gfx1250) — compile-verified
//
#include <hip/hip_runtime.h>
#include <stddef.h>

// LightGCN-style propagation for MI455X (gfx1250, wave32).
// Edge phase: 1 wave per edge, lane owns 16B of the 128-float row,
//             scatter via no-return global_atomic_add_f32 (L2 atomic units;
//             the 153MB embedding tables are L2-resident on the 192MB L2).
// LN phase:   1 wave per 16 rows, row stats via V_WMMA_F32_16X16X4_F32.

typedef __attribute__((ext_vector_type(2))) float v2f;
typedef __attribute__((ext_vector_type(8))) float v8f;

#define LN_EPS 1e-5f

__device__ __forceinline__ void atomic_add_f32(float* p, float v) {
    // relaxed, device scope, result discarded -> global_atomic_add_f32 (no-return)
    __hip_atomic_fetch_add(p, v, __ATOMIC_RELAXED, __HIP_MEMORY_SCOPE_AGENT);
}

__global__ void copy_f4_kernel(float4* __restrict__ dst,
                               const float4* __restrict__ src, long n4) {
    long i = (long)blockIdx.x * blockDim.x + threadIdx.x;
    if (i < n4) dst[i] = src[i];
}

// One wave32 per edge. Lane l handles floats [4l, 4l+4) of the 128-wide row.
// acc_u[r] += v * cur_i[c];  acc_i[c] += v * cur_u[r]
__global__ void edge_scatter_kernel(const int* __restrict__ rows,
                                    const int* __restrict__ cols,
                                    const float* __restrict__ vals,
                                    const float* __restrict__ cu,
                                    const float* __restrict__ ci,
                                    float* __restrict__ au,
                                    float* __restrict__ ai,
                                    int nedges) {
    int wave = blockIdx.x * (blockDim.x >> 5) + (threadIdx.x >> 5);
    if (wave >= nedges) return;
    int lane = threadIdx.x & 31;

    // wave-uniform loads; non-temporal: edge lists are streamed once and must
    // not evict the L2-resident embedding tables
    int   r = __builtin_nontemporal_load(rows + wave);
    int   c = __builtin_nontemporal_load(cols + wave);
    float v = __builtin_nontemporal_load(vals + wave);

    size_t offu = (size_t)r * 128 + (size_t)(lane << 2);
    size_t offi = (size_t)c * 128 + (size_t)(lane << 2);

    const float4 xu = *(const float4*)(cu + offu);   // global_load_b128
    const float4 xi = *(const float4*)(ci + offi);

    float* pu = au + offu;
    float* pi = ai + offi;
    atomic_add_f32(pu + 0, v * xi.x);
    atomic_add_f32(pu + 1, v * xi.y);
    atomic_add_f32(pu + 2, v * xi.z);
    atomic_add_f32(pu + 3, v * xi.w);
    atomic_add_f32(pi + 0, v * xu.x);
    atomic_add_f32(pi + 1, v * xu.y);
    atomic_add_f32(pi + 2, v * xu.z);
    atomic_add_f32(pi + 3, v * xu.w);
}

// LayerNorm over rows of 128. One wave per 16-row tile.
// Row sums / sum-of-squares computed on the matrix pipe:
//   D += A(16x4 chunk of x) * B(ones)  via V_WMMA_F32_16X16X4_F32, 32 chunks.
// f32 A 16x4 layout: lanes 0-15 hold (row=lane, K=0,1), lanes 16-31 (row=lane-16, K=2,3).
// f32 C/D 16x16 layout: VGPR j holds row j (lanes 0-15) / row j+8 (lanes 16-31);
// with B=ones every column of D equals the row sum, so each lane has 8 row sums.
// Optional second destination y2: used after layer 0 so the layer-1
// accumulator is initialized for free (skips a 2x153MB copy pass).
__global__ void ln_wmma_kernel(const float* __restrict__ x,
                               float* __restrict__ y,
                               float* __restrict__ y2,
                               const float* __restrict__ gamma,
                               const float* __restrict__ beta,
                               int ntiles, int nrows) {
    int wave = blockIdx.x * (blockDim.x >> 5) + (threadIdx.x >> 5);
    if (wave >= ntiles) return;                  // wave-uniform: EXEC stays all-ones
    int lane = threadIdx.x & 31;
    int half = lane >> 4;                        // 0: K=0,1  1: K=2,3
    int m    = lane & 15;
    int row0 = wave << 4;

    int rload = row0 + m;
    if (rload > nrows - 1) rload = nrows - 1;    // clamp (tiles are exact here)
    const float* base = x + (size_t)rload * 128 + (size_t)(half << 1);

    v8f s = {};                                  // row sums
    v8f q = {};                                  // row sums of squares
    v2f ones;
    ones[0] = 1.0f; ones[1] = 1.0f;

#pragma unroll
    for (int c = 0; c < 32; ++c) {
        v2f a = *(const v2f*)(base + c * 4);     // A chunk: this row, K=4c+2*half+{0,1}
        s = __builtin_amdgcn_wmma_f32_16x16x4_f32(false, a, false, ones,
                                                  (short)0, s, false, false);
        v2f aa = a * a;
        q = __builtin_amdgcn_wmma_f32_16x16x4_f32(false, aa, false, ones,
                                                  (short)0, q, false, false);
    }

    // This lane writes columns [8m, 8m+8) of rows row0 + half*8 + {0..7}
    const float4 g0 = *(const float4*)(gamma + m * 8);
    const float4 g1 = *(const float4*)(gamma + m * 8 + 4);
    const float4 b0 = *(const float4*)(beta + m * 8);
    const float4 b1 = *(const float4*)(beta + m * 8 + 4);

#pragma unroll
    for (int j = 0; j < 8; ++j) {
        int r = row0 + j + (half << 3);          // matches C/D VGPR-j row mapping
        if (r >= nrows) continue;
        float mean = s[j] * (1.0f / 128.0f);
        float var  = q[j] * (1.0f / 128.0f) - mean * mean;
        float rstd = rsqrtf(var + LN_EPS);

        const float* xr = x + (size_t)r * 128 + m * 8;
        size_t       wo = (size_t)r * 128 + m * 8;
        float4 x0 = *(const float4*)xr;
        float4 x1 = *(const float4*)(xr + 4);
        float4 y0, y1;
        y0.x = g0.x * ((x0.x - mean) * rstd) + b0.x;
        y0.y = g0.y * ((x0.y - mean) * rstd) + b0.y;
        y0.z = g0.z * ((x0.z - mean) * rstd) + b0.z;
        y0.w = g0.w * ((x0.w - mean) * rstd) + b0.w;
        y1.x = g1.x * ((x1.x - mean) * rstd) + b1.x;
        y1.y = g1.y * ((x1.y - mean) * rstd) + b1.y;
        y1.z = g1.z * ((x1.z - mean) * rstd) + b1.z;
        y1.w = g1.w * ((x1.w - mean) * rstd) + b1.w;
        *(float4*)(y + wo)     = y0;
        *(float4*)(y + wo + 4) = y1;
        if (y2) {                                // wave-uniform predicate
            *(float4*)(y2 + wo)     = y0;
            *(float4*)(y2 + wo + 4) = y1;
        }
    }
}

extern "C" void kernel_launch(void* const* d_in, const int* in_sizes, int n_in,
                              void* d_out, int out_size, void* d_ws, size_t ws_size,
                              hipStream_t stream) {
    (void)n_in; (void)out_size; (void)ws_size;

    const float* user_emb = (const float*)d_in[0];
    const float* item_emb = (const float*)d_in[1];
    const float* vals     = (const float*)d_in[2];
    const float* gamma    = (const float*)d_in[3];
    const float* beta     = (const float*)d_in[4];
    const int*   rows     = (const int*)d_in[5];
    const int*   cols     = (const int*)d_in[6];

    const long n_u    = (long)in_sizes[0];       // user floats (rows * 128)
    const long n_i    = (long)in_sizes[1];       // item floats
    const int  nedges = in_sizes[2];
    const long total  = n_u + n_i;
    const int  nrows  = (int)(total / 128);
    const int  ntiles = (nrows + 15) / 16;

    float* out = (float*)d_out;                  // [u_emb ; i_emb] accumulator/result
    float* ws  = (float*)d_ws;                   // inter-layer embedding buffer

    const int CT = 256;                          // 8 waves / block
    const int edge_blocks = (nedges + 7) / 8;
    const int ln_blocks   = (ntiles + 7) / 8;
    long n4u = n_u / 4, n4i = n_i / 4;

    // ---- layer 0: acc(out) = input emb; scatter; LN -> ws AND out ----
    copy_f4_kernel<<<(int)((n4u + CT - 1) / CT), CT, 0, stream>>>(
        (float4*)out, (const float4*)user_emb, n4u);
    copy_f4_kernel<<<(int)((n4i + CT - 1) / CT), CT, 0, stream>>>(
        (float4*)(out + n_u), (const float4*)item_emb, n4i);
    edge_scatter_kernel<<<edge_blocks, CT, 0, stream>>>(
        rows, cols, vals, user_emb, item_emb, out, out + n_u, nedges);
    // dual store: ws = layer-1 gather source, out = layer-1 accumulator init
    ln_wmma_kernel<<<ln_blocks, CT, 0, stream>>>(
        out, ws, out, gamma, beta, ntiles, nrows);

    // ---- layer 1: scatter from ws into out (== LN(layer0) already); LN in place ----
    edge_scatter_kernel<<<edge_blocks, CT, 0, stream>>>(
        rows, cols, vals, ws, ws + n_u, out, out + n_u, nedges);
    ln_wmma_kernel<<<ln_blocks, CT, 0, stream>>>(
        out, out, nullptr, gamma, beta, ntiles, nrows);
}